// DTWLAYER_SHORTKERNEL_38448547233964
// MI455X (gfx1250) — compile-verified
//
#include <hip/hip_runtime.h>

// Problem constants (from the reference)
#define C_ 8
#define L_ 4096
#define F_ 32
#define K_ 10
#define PROC_ 20
#define STEP_ 5
#define NW_ 815                 // len(range(0, 4096-20-5+1, 5)); CHAN_OUT == NW -> no pad region

// Block shape: 32 windows (lanes) x 8 kernels (waves) = 256 threads
#define WIN_T 32
#define FW 8
#define WTILES ((NW_ + WIN_T - 1) / WIN_T)          // 26
#define XTILE 176                                    // (WIN_T-1)*STEP + PROC = 175, rounded to x4
#define XLOADS (XTILE / 4)                           // 44 b128 copies
#define KLOADS ((FW * K_) / 4)                       // 20 b128 copies (80 floats)

// Low 32 bits of a generic pointer to __shared__ == wave-relative LDS byte address.
__device__ __forceinline__ unsigned lds_byte_addr(const void* p) {
    return (unsigned)(unsigned long long)(size_t)p;
}

// CDNA5 async global->LDS copy, 16 bytes, tracked by ASYNCcnt.
__device__ __forceinline__ void async_copy_b128(unsigned lds_addr, const void* gptr) {
    asm volatile("global_load_async_to_lds_b128 %0, %1, off"
                 :
                 : "v"(lds_addr), "v"((unsigned long long)(size_t)gptr)
                 : "memory");
}

__global__ __launch_bounds__(256) void
DTWLAYER_SHORTKERNEL_38448547233964_kernel(const float* __restrict__ x,
                                           const float* __restrict__ kernels,
                                           float* __restrict__ out) {
    __shared__ __align__(16) float sX[XTILE];        // 32 overlapping windows
    __shared__ __align__(16) float sKern[FW * K_];   // this block's 8 kernels

    const int tid = threadIdx.x;
    const int b   = blockIdx.z;
    const int yc  = blockIdx.y;           // = c * (F/FW) + fg
    const int c   = yc >> 2;
    const int fg  = yc & 3;               // kernel group (0..3)
    const int w0  = blockIdx.x * WIN_T;   // first window of this block
    const int s0  = w0 * STEP_;           // first sample of the x tile (s0*4 is 16B aligned)

    // ---- Stage data into LDS via CDNA5 async global->LDS path ----
    if (tid < XLOADS) {
        const int off = tid * 4;
        if (s0 + off < L_) {              // clamp for the last window tile
            const float* g = x + ((size_t)(b * C_ + c)) * L_ + s0 + off;
            async_copy_b128(lds_byte_addr(&sX[0]) + tid * 16, g);
        }
    }
    if (tid >= 64 && tid < 64 + KLOADS) { // 80 kernel floats at 320B-aligned offset
        const int t = tid - 64;
        async_copy_b128(lds_byte_addr(&sKern[0]) + t * 16,
                        kernels + (size_t)fg * (FW * K_) + t * 4);
    }
    asm volatile("s_wait_asynccnt 0" ::: "memory");  // my wave's async copies done
    __syncthreads();                                 // all waves' copies visible

    // ---- Mapping: lane = window (coalesced stores), wave = kernel (uniform kv) ----
    const int lane  = tid & 31;           // window-in-tile
    const int wavef = tid >> 5;           // kernel-in-group (uniform per wave)
    const int f     = fg * FW + wavef;    // global kernel index
    const int w     = w0 + lane;          // global window index

    float win[PROC_];
#pragma unroll
    for (int j = 0; j < PROC_; ++j) win[j] = sX[lane * STEP_ + j];   // stride-5: bank-conflict-free

    float kv[K_];
#pragma unroll
    for (int i = 0; i < K_; ++i) kv[i] = sKern[wavef * K_ + i];      // wave-uniform

    // ---- DTW DP, all in registers (exact reference recurrence) ----
    float row[PROC_];
    float a = 0.0f;
#pragma unroll
    for (int j = 0; j < PROC_; ++j) {                 // row0 = cumsum(D[0])
        const float d = kv[0] - win[j];
        a = fmaf(d, d, a);
        row[j] = a;
    }
#pragma unroll
    for (int i = 1; i < K_; ++i) {
        float diag = row[0];                          // prev_row[0]
        const float d0 = kv[i] - win[0];
        float cur = fmaf(d0, d0, diag);               // col 0: only "up"
        row[0] = cur;
        float left = cur;
#pragma unroll
        for (int j = 1; j < PROC_; ++j) {
            const float up = row[j];
            const float d  = kv[i] - win[j];
            const float m  = fminf(fminf(left, up), diag);
            cur = fmaf(d, d, m);
            diag = up;
            row[j] = cur;
            left = cur;
        }
    }

    // out[b, c*F + f, w] — 32 consecutive lanes -> one contiguous 128B store per wave
    if (w < NW_) {
        out[((size_t)((b * C_ + c) * F_ + f)) * NW_ + w] = row[PROC_ - 1];
    }
}

extern "C" void kernel_launch(void* const* d_in, const int* in_sizes, int n_in,
                              void* d_out, int out_size, void* d_ws, size_t ws_size,
                              hipStream_t stream) {
    (void)n_in; (void)d_ws; (void)ws_size; (void)out_size;
    const float* x       = (const float*)d_in[0];   // (B, C, L) f32
    const float* kernels = (const float*)d_in[1];   // (F, K)    f32
    float*       out     = (float*)d_out;           // (B, C*F, NW) f32

    const int B = in_sizes[0] / (C_ * L_);          // 8 for the reference setup
    dim3 grid(WTILES, (F_ / FW) * C_, B);           // (26, 32, 8)
    DTWLAYER_SHORTKERNEL_38448547233964_kernel<<<grid, 256, 0, stream>>>(x, kernels, out);
}